// FixedHiddenMlp_34875134444033
// MI455X (gfx1250) — compile-verified
//
#include <hip/hip_runtime.h>
#include <math.h>

// MI455X / gfx1250, wave32. Fused 3-GEMM MLP chain, F=20, fp32 WMMA 16x16x4.
// Memory-bound: 84 MB of x @ 23.3 TB/s ~ 3.6us floor; weights live in VGPRs.
// This revision removes per-load exec predication (clamped addressing) and
// software-pipelines the x-tile loads across loop iterations so HBM latency
// overlaps the 3-GEMM + 2-LDS-bounce compute chain.

typedef __attribute__((ext_vector_type(2))) float v2f;
typedef __attribute__((ext_vector_type(8))) float v8f;

#define FDIM 20
#define KCH  5          // 5 chunks of K=4 cover K=20 exactly
#define LDSW 34         // padded LDS row stride (even -> 8B-aligned b64 LDS loads)
#define WAVES_PER_BLOCK 8
#define BLOCK_THREADS   256
#define NBLOCKS         2048

static __device__ __forceinline__ v8f wmma_f32(v2f a, v2f b, v8f c) {
    return __builtin_amdgcn_wmma_f32_16x16x4_f32(
        /*neg_a=*/false, a, /*neg_b=*/false, b,
        /*c_mod=*/(short)0, c, /*reuse_a=*/false, /*reuse_b=*/false);
}

__global__ __launch_bounds__(BLOCK_THREADS)
void mlp_fused_kernel(const float* __restrict__ x,
                      const float* __restrict__ W,     // [20][20] (out, in)
                      const float* __restrict__ bvec,  // [20]
                      const float* __restrict__ R,     // [20][20] rand_weight
                      float* __restrict__ partials,    // [gridDim.x][2] = {sum, sumAbs}
                      int N)                           // rows
{
    __shared__ float hbuf[WAVES_PER_BLOCK][16 * LDSW];
    __shared__ float red[WAVES_PER_BLOCK][2];

    const int tid  = threadIdx.x;
    const int wave = tid >> 5;
    const int lane = tid & 31;
    const int ln   = lane & 15;   // row (A) / col (B,D) sub-index
    const int hi   = lane >> 4;   // half-wave select

    // ---------------- Resident B fragments (built once per wave) -------------
    // B layout (4x16 f32): VGPR0 holds K=k0 (lanes 0-15) / K=k0+2 (lanes 16-31),
    // VGPR1 holds K=k0+1 / K=k0+3 -> per-lane v2f = {B[k0+2hi][n], B[k0+2hi+1][n]}.
    v2f Bw[2][KCH];   // GEMM1/3: B[k][n] = W[n][k]   (x @ W^T)
    v2f Br[2][KCH];   // GEMM2  : B[k][n] = R[k][n]   (h @ R)
    float bias[2];
    #pragma unroll
    for (int t = 0; t < 2; ++t) {
        const int n = ln + 16 * t;          // output column 0..31
        const bool valid = (n < FDIM);
        bias[t] = valid ? bvec[n] : 0.0f;
        #pragma unroll
        for (int kc = 0; kc < KCH; ++kc) {
            const int k0 = kc * 4 + 2 * hi;
            v2f w = {0.0f, 0.0f}, r = {0.0f, 0.0f};
            if (valid) {
                w = *(const v2f*)(W + n * FDIM + k0);    // W[n][k0], W[n][k0+1]
                r.x = R[(k0 + 0) * FDIM + n];
                r.y = R[(k0 + 1) * FDIM + n];
            }
            Bw[t][kc] = w;
            Br[t][kc] = r;
        }
    }

    float* myh = hbuf[wave];
    float lsum = 0.0f, labs = 0.0f;

    const int nRowTiles  = (N + 15) >> 4;
    const int wavesTotal = (gridDim.x * blockDim.x) >> 5;
    const int gwave      = (blockIdx.x * blockDim.x + tid) >> 5;

    // Unpredicated clamped fragment load: no exec juggling, loads batch freely.
    // Phantom rows (row >= N) read row 0; excluded later by the row guard.
    auto load_frag = [&](int tile, v2f (&a)[KCH]) {
        int row = tile * 16 + ln;
        row = (row < N) ? row : 0;
        row = (row < 0) ? 0 : row;
        const float* xr = x + (size_t)row * FDIM;
        #pragma unroll
        for (int kc = 0; kc < KCH; ++kc)
            a[kc] = *(const v2f*)(xr + kc * 4 + 2 * hi);   // 8B aligned b64 loads
    };

    v2f acur[KCH];
    if (gwave < nRowTiles) load_frag(gwave, acur);

    for (int tile = gwave; tile < nRowTiles; tile += wavesTotal) {
        // -------- Prefetch next tile (overlaps the whole 3-GEMM chain) -------
        v2f anext[KCH];
        load_frag(tile + wavesTotal, anext);   // clamped; harmless on last iter

        // ---------------- GEMM1: h1 = x @ W^T + b ----------------------------
        v8f d0 = {}; v8f d1 = {};
        #pragma unroll
        for (int kc = 0; kc < KCH; ++kc) {
            d0 = wmma_f32(acur[kc], Bw[0][kc], d0);
            d1 = wmma_f32(acur[kc], Bw[1][kc], d1);
        }
        // D layout: VGPR v -> row M = v + 8*hi, col = ln (+16 for tile 1).
        #pragma unroll
        for (int v = 0; v < 8; ++v) {
            const int r_ = v + 8 * hi;
            myh[r_ * LDSW + ln +  0] = d0[v] + bias[0];
            myh[r_ * LDSW + ln + 16] = d1[v] + bias[1];  // cols>=20 stay 0
        }
        asm volatile("s_wait_dscnt 0" ::: "memory");     // wave-local LDS fence

        // ---------------- GEMM2: h2 = relu(h1 @ R + 1) -----------------------
        v8f e0 = {}; v8f e1 = {};
        #pragma unroll
        for (int kc = 0; kc < KCH; ++kc) {
            v2f a = *(const v2f*)(&myh[ln * LDSW + kc * 4 + 2 * hi]);
            e0 = wmma_f32(a, Br[0][kc], e0);
            e1 = wmma_f32(a, Br[1][kc], e1);
        }
        const bool col1ok = (ln + 16 < FDIM);            // predicate: +1 pollutes pad
        #pragma unroll
        for (int v = 0; v < 8; ++v) {
            const int r_ = v + 8 * hi;
            myh[r_ * LDSW + ln +  0] = fmaxf(e0[v] + 1.0f, 0.0f);
            myh[r_ * LDSW + ln + 16] = col1ok ? fmaxf(e1[v] + 1.0f, 0.0f) : 0.0f;
        }
        asm volatile("s_wait_dscnt 0" ::: "memory");

        // ---------------- GEMM3: h3 = h2 @ W^T + b, then reduce ---------------
        v8f f0 = {}; v8f f1 = {};
        #pragma unroll
        for (int kc = 0; kc < KCH; ++kc) {
            v2f a = *(const v2f*)(&myh[ln * LDSW + kc * 4 + 2 * hi]);
            f0 = wmma_f32(a, Bw[0][kc], f0);
            f1 = wmma_f32(a, Bw[1][kc], f1);
        }
        if (tile * 16 + 16 <= N) {
            // Full tile (common case): unguarded accumulation.
            #pragma unroll
            for (int v = 0; v < 8; ++v) {
                const float g0 = f0[v] + bias[0];
                const float g1 = f1[v] + bias[1];        // cols>=20 exact zero
                lsum += g0 + g1;
                labs += fabsf(g0) + fabsf(g1);
            }
        } else {
            // Tail tile: drop phantom pad rows.
            #pragma unroll
            for (int v = 0; v < 8; ++v) {
                if (tile * 16 + v + 8 * hi < N) {
                    const float g0 = f0[v] + bias[0];
                    const float g1 = f1[v] + bias[1];
                    lsum += g0 + g1;
                    labs += fabsf(g0) + fabsf(g1);
                }
            }
        }

        // -------- Rotate pipeline ---------------------------------------------
        #pragma unroll
        for (int kc = 0; kc < KCH; ++kc) acur[kc] = anext[kc];
    }

    // ---------------- Deterministic reduction --------------------------------
    #pragma unroll
    for (int off = 16; off > 0; off >>= 1) {
        lsum += __shfl_xor(lsum, off, 32);
        labs += __shfl_xor(labs, off, 32);
    }
    if (lane == 0) { red[wave][0] = lsum; red[wave][1] = labs; }
    __syncthreads();
    if (tid == 0) {
        float s0 = 0.0f, s1 = 0.0f;
        #pragma unroll
        for (int wv = 0; wv < WAVES_PER_BLOCK; ++wv) { s0 += red[wv][0]; s1 += red[wv][1]; }
        partials[2 * blockIdx.x + 0] = s0;
        partials[2 * blockIdx.x + 1] = s1;
    }
}

__global__ __launch_bounds__(256)
void finalize_kernel(const float* __restrict__ partials, int nBlocks,
                     float* __restrict__ out)
{
    __shared__ float s[256][2];
    const int t = threadIdx.x;
    float a = 0.0f, c = 0.0f;
    for (int i = t; i < nBlocks; i += 256) {          // fixed-order strided accum
        a += partials[2 * i + 0];
        c += partials[2 * i + 1];
    }
    s[t][0] = a; s[t][1] = c;
    __syncthreads();
    for (int off = 128; off > 0; off >>= 1) {         // fixed tree -> deterministic
        if (t < off) { s[t][0] += s[t + off][0]; s[t][1] += s[t + off][1]; }
        __syncthreads();
    }
    if (t == 0) {
        const float total = s[0][0];
        const float sabs  = s[0][1];
        // k = smallest k>=0 with sabs * 2^-k <= 1  (mirrors reference guards)
        int k = (int)ceilf(log2f(fmaxf(sabs, 1.0f)));
        if (ldexpf(sabs, -k) > 1.0f) k += 1;
        if (k > 0 && ldexpf(sabs, -(k - 1)) <= 1.0f) k -= 1;
        out[0] = ldexpf(total, -k);
    }
}

extern "C" void kernel_launch(void* const* d_in, const int* in_sizes, int n_in,
                              void* d_out, int out_size, void* d_ws, size_t ws_size,
                              hipStream_t stream) {
    const float* x = (const float*)d_in[0];   // [N][20]
    const float* W = (const float*)d_in[1];   // [20][20]
    const float* b = (const float*)d_in[2];   // [20]
    const float* R = (const float*)d_in[3];   // [20][20]
    float* out      = (float*)d_out;          // scalar
    float* partials = (float*)d_ws;           // [NBLOCKS][2], fully rewritten each call

    const int N = in_sizes[0] / FDIM;

    mlp_fused_kernel<<<NBLOCKS, BLOCK_THREADS, 0, stream>>>(x, W, b, R, partials, N);
    finalize_kernel<<<1, 256, 0, stream>>>(partials, NBLOCKS, out);
}